// Crf_27462020890809
// MI455X (gfx1250) — compile-verified
//
#include <hip/hip_runtime.h>
#include <stdint.h>

// CRF Viterbi decode for MI455X (gfx1250, wave32).
// B=1024 chains, T=512 steps, C=41 tags.
// - One wave per chain; lane L owns tags L and L+32.
// - Transition columns live in 82 VGPRs (loop-invariant over T).
// - Score vector lives in registers across lanes; broadcast via v_readlane.
// - Emission rows streamed with double-buffered GLOBAL_LOAD_ASYNC_TO_LDS_B32
//   (ASYNCcnt pipeline), so the serial T-recurrence never waits on HBM.
// - Backtrace staged in LDS, then written to global coalesced.
// WMMA is intentionally absent: Viterbi is a (max,+) tropical-semiring
// recurrence with argmax side outputs; v_wmma only implements (*,+).

#define B_ 1024
#define T_ 512
#define C_ 41

// --- async global->LDS prefetch of one emission row (41 floats) -------------
// Two b32 async ops: elements 0..31 (all lanes) and 32..40 (lanes 0..8;
// remaining lanes re-load element 0 into a dummy LDS slot to stay in-bounds).
__device__ __forceinline__ void async_prefetch_row(const float* grow,
                                                   unsigned lds_base,
                                                   int lane) {
    unsigned lds0 = lds_base + (unsigned)lane * 4u;
    const float* g0 = grow + lane;
    asm volatile("global_load_async_to_lds_b32 %0, %1, off"
                 :: "v"(lds0), "v"(g0) : "memory");

    bool v = lane < (C_ - 32);                       // lanes 0..8 valid
    unsigned lds1 = lds_base + (v ? (32u + (unsigned)lane) * 4u : 44u * 4u);
    const float* g1 = grow + (v ? (32 + lane) : 0);
    asm volatile("global_load_async_to_lds_b32 %0, %1, off"
                 :: "v"(lds1), "v"(g1) : "memory");
}

__device__ __forceinline__ float bcast_lane(float v, int srclane) {
    return __int_as_float(__builtin_amdgcn_readlane(__float_as_int(v), srclane));
}

__global__ __launch_bounds__(32) void crf_viterbi_kernel(
    const float* __restrict__ x,        // [B, T, C]
    const float* __restrict__ start_t,  // [C]
    const float* __restrict__ end_t,    // [C]
    const float* __restrict__ trans,    // [C, C]
    long long* __restrict__ out,        // [B, T] int64 tags
    unsigned char* __restrict__ bp)     // [T-1, B, C] backpointers (d_ws)
{
    const int b    = blockIdx.x;
    const int lane = threadIdx.x;                 // 0..31
    const int j0   = lane;                        // tag 0..31
    const bool hasJ1 = (lane < (C_ - 32));        // lanes 0..8 own tags 32..40
    const int j1   = hasJ1 ? (lane + 32) : (C_ - 1);  // clamp (results masked)

    __shared__ float embuf[2][48];        // double-buffered emission row (+pad)
    __shared__ unsigned char tagbuf[T_];  // decoded path staging

    // ---- transition columns -> registers (loop-invariant over T) ----------
    float trA[C_], trB[C_];
    #pragma unroll
    for (int i = 0; i < C_; ++i) {
        trA[i] = trans[i * C_ + j0];
        trB[i] = trans[i * C_ + j1];
    }

    const size_t xbase = (size_t)b * T_ * C_;

    // ---- t = 0 init --------------------------------------------------------
    float sc0 = start_t[j0] + x[xbase + j0];
    float sc1 = start_t[j1] + x[xbase + j1];      // lanes 0..8 meaningful

    const unsigned emb0 = (unsigned)(uintptr_t)&embuf[0][0];
    const unsigned emb1 = (unsigned)(uintptr_t)&embuf[1][0];

    // prefetch emissions for t = 1 into buffer 1
    async_prefetch_row(x + xbase + C_, emb1, lane);

    // ---- forward recurrence ------------------------------------------------
    for (int t = 1; t < T_; ++t) {
        // wait for the async fill of buf[t&1] issued last iteration
        asm volatile("s_wait_asynccnt 0x0" ::: "memory");

        float e0 = embuf[t & 1][j0];
        float e1 = embuf[t & 1][j1];
        // our DS reads of this buffer must retire before it can be refilled
        asm volatile("s_wait_dscnt 0x0" ::: "memory");

        if (t + 1 < T_)
            async_prefetch_row(x + xbase + (size_t)(t + 1) * C_,
                               (t & 1) ? emb0 : emb1, lane);

        // max-plus over previous tags i = 0..40; bit-faithful to the ref:
        // cand = (score[i] + trans[i][j]) + emit[j]; strict > keeps the
        // first-occurrence argmax like jnp.argmax.
        float m0, m1; int a0, a1;
        {
            float s = bcast_lane(sc0, 0);
            m0 = (s + trA[0]) + e0; a0 = 0;
            m1 = (s + trB[0]) + e1; a1 = 0;
        }
        #pragma unroll
        for (int i = 1; i < C_; ++i) {
            float sv = (i < 32) ? sc0 : sc1;
            float s  = bcast_lane(sv, i & 31);
            float c0 = (s + trA[i]) + e0;
            if (c0 > m0) { m0 = c0; a0 = i; }
            float c1 = (s + trB[i]) + e1;
            if (c1 > m1) { m1 = c1; a1 = i; }
        }
        sc0 = m0;
        sc1 = m1;

        const size_t bpo = ((size_t)(t - 1) * B_ + b) * C_;
        bp[bpo + j0] = (unsigned char)a0;
        if (hasJ1) bp[bpo + j1] = (unsigned char)a1;
    }

    // ---- final scores + argmax over tags (pure readlane, no LDS) ----------
    float f0 = sc0 + end_t[j0];
    float f1 = sc1 + end_t[j1];
    float bestv = bcast_lane(f0, 0);
    int   bestj = 0;
    #pragma unroll
    for (int j = 1; j < C_; ++j) {
        float v = bcast_lane((j < 32) ? f0 : f1, j & 31);
        if (v > bestv) { bestv = v; bestj = j; }
    }

    // ---- backtrace (serial chain; bp is L2-resident: 21.5 MB << 192 MB) ---
    if (lane == 0) {
        int tag = bestj;
        tagbuf[T_ - 1] = (unsigned char)tag;
        for (int t = T_ - 2; t >= 0; --t) {
            tag = (int)bp[((size_t)t * B_ + b) * C_ + tag];
            tagbuf[t] = (unsigned char)tag;
        }
    }
    __syncthreads();   // single-wave WG: barrier -> NOP, but orders LDS

    // ---- coalesced int64 output: 16 wave-wide b64 stores ------------------
    long long* ob = out + (size_t)b * T_;
    #pragma unroll
    for (int k = 0; k < T_ / 32; ++k) {
        int idx = k * 32 + lane;
        ob[idx] = (long long)tagbuf[idx];
    }
}

extern "C" void kernel_launch(void* const* d_in, const int* in_sizes, int n_in,
                              void* d_out, int out_size, void* d_ws, size_t ws_size,
                              hipStream_t stream) {
    (void)in_sizes; (void)n_in; (void)out_size;
    const float* x     = (const float*)d_in[0];   // [B,T,C] f32
    const float* start = (const float*)d_in[1];   // [C]
    const float* endt  = (const float*)d_in[2];   // [C]
    const float* trans = (const float*)d_in[3];   // [C,C]
    long long* out = (long long*)d_out;           // [B,T] int64
    unsigned char* bp = (unsigned char*)d_ws;     // needs (T-1)*B*C = ~21.5 MB
    (void)ws_size;

    crf_viterbi_kernel<<<B_, 32, 0, stream>>>(x, start, endt, trans, out, bp);
}